// NMSDecoder_3607772529382
// MI455X (gfx1250) — compile-verified
//
#include <hip/hip_runtime.h>
#include <cstdint>
#include <cstddef>

// Problem constants (fixed by the reference).
#define NVARS   8192
#define NCHECKS 4096
#define DCdeg   6
#define DVdeg   3
#define NEDGES  (NCHECKS * DCdeg)   // 24576
#define NBATCH  2048
#define NITER   5
#define CBATCH  512                 // batch chunk -> 80MB working set, L2-resident
#define RS4     (CBATCH / 4)        // row stride in float4 units

typedef __attribute__((ext_vector_type(4))) unsigned int u32x4;
typedef __attribute__((ext_vector_type(8))) int          i32x8;
typedef __attribute__((ext_vector_type(4))) int          i32x4;

// ---------------------------------------------------------------- graph setup
__global__ void k_zero_cnt(int* cnt) {
    int i = blockIdx.x * 256 + threadIdx.x;
    if (i < NVARS) cnt[i] = 0;
}

__global__ void k_build_v2e(const int* __restrict__ Hc, int* cnt, int* v2e) {
    int e = blockIdx.x * 256 + threadIdx.x;
    if (e < NEDGES) {
        int v = Hc[e];
        int s = atomicAdd(cnt + v, 1);
        v2e[v * DVdeg + s] = e;
    }
}

// Sort each variable's 3 edges ascending -> deterministic FP summation order.
__global__ void k_sort_v2e(int* v2e) {
    int v = blockIdx.x * 256 + threadIdx.x;
    if (v < NVARS) {
        int a = v2e[v * 3], b = v2e[v * 3 + 1], c = v2e[v * 3 + 2], t;
        if (a > b) { t = a; a = b; b = t; }
        if (b > c) { t = b; b = c; c = t; }
        if (a > b) { t = a; a = b; b = t; }
        v2e[v * 3] = a; v2e[v * 3 + 1] = b; v2e[v * 3 + 2] = c;
    }
}

// ------------------------------------------- transpose-in: ch[B][V] -> chT[V][CB]
// TDM loads the 32(b) x 32(v) tile straight into LDS (2-DWORD pad per 32 DWORDs
// -> row stride 34 floats), then all 8 waves write the transposed tile coalesced.
__launch_bounds__(256)
__global__ void k_transpose_in(const float* __restrict__ ch, float* __restrict__ chT,
                               int b0_global) {
    __shared__ float tile[32 * 34];
    const int jb0 = blockIdx.x * 32;         // local batch base within chunk
    const int v0  = blockIdx.y * 32;

    if (threadIdx.x < 32) {                  // wave 0 issues one TDM descriptor
        uint64_t ga = (uint64_t)(uintptr_t)ch +
                      ((uint64_t)(b0_global + jb0) * NVARS + (uint64_t)v0) * 4u;
        unsigned lds = (unsigned)(uintptr_t)(&tile[0]);
        u32x4 g0 = {};
        g0[0] = 1u;                                    // count=1, user descriptor
        g0[1] = lds;                                   // lds_addr
        g0[2] = (unsigned)ga;                          // global_addr[31:0]
        g0[3] = (unsigned)(ga >> 32) | 0x80000000u;    // addr[56:32] | type=2
        i32x8 g1 = {};
        //        data_size=4B  pad_en   interval=32dw  amount=2dw
        g1[0] = (2 << 16) | (1 << 20) | (4 << 22) | (1 << 25);
        g1[1] = (int)((NVARS & 0xFFFF) << 16);         // tensor_dim0 lo
        g1[2] = (int)(((NVARS >> 16) & 0xFFFF) | ((NBATCH & 0xFFFF) << 16));
        g1[3] = (int)(((NBATCH >> 16) & 0xFFFF) | (32 << 16));  // tile_dim0=32
        g1[4] = 32;                                    // tile_dim1=32, tile_dim2=0
        g1[5] = NVARS;                                 // tensor_dim0_stride
        i32x4 gz = {};
#if defined(__clang_major__) && (__clang_major__ >= 23)
        i32x8 gz8 = {};
        __builtin_amdgcn_tensor_load_to_lds(g0, g1, gz, gz, gz8, 0);
#else
        __builtin_amdgcn_tensor_load_to_lds(g0, g1, gz, gz, 0);
#endif
        __builtin_amdgcn_s_wait_tensorcnt(0);
    }
    __syncthreads();

    const int bl = threadIdx.x & 31;         // lane = batch -> coalesced store
    for (int vv = threadIdx.x >> 5; vv < 32; vv += 8) {
        chT[(size_t)(v0 + vv) * CBATCH + jb0 + bl] = tile[bl * 34 + vv];
    }
}

// ------------------------- variable-node update pass (float4 over the batch dim)
// One 128-thread block covers a full 512-column chunk row with b128 accesses.
__launch_bounds__(128)
__global__ void k_var(const float* __restrict__ chT, const float* __restrict__ c2v,
                      const int* __restrict__ v2e, float* __restrict__ varllr) {
    const int j4 = threadIdx.x;
    const int v  = blockIdx.y;
    const int e0 = v2e[v * 3], e1 = v2e[v * 3 + 1], e2 = v2e[v * 3 + 2];
    const float4* c4  = (const float4*)c2v;
    const float4* ch4 = (const float4*)chT;
    float4*       vl4 = (float4*)varllr;

    float4 a0 = c4[(size_t)e0 * RS4 + j4];
    float4 a1 = c4[(size_t)e1 * RS4 + j4];
    float4 a2 = c4[(size_t)e2 * RS4 + j4];
    float4 cv = ch4[(size_t)v * RS4 + j4];
    float4 o;
    o.x = cv.x + (a0.x + a1.x + a2.x);
    o.y = cv.y + (a0.y + a1.y + a2.y);
    o.z = cv.z + (a0.z + a1.z + a2.z);
    o.w = cv.w + (a0.w + a1.w + a2.w);
    vl4[(size_t)v * RS4 + j4] = o;
}

// -------------------- check-node update (weighted min-sum, dc = 6, float4 batch)
__launch_bounds__(128)
__global__ void k_check(const float* __restrict__ varllr, float* __restrict__ c2v,
                        const int* __restrict__ Hc, const float* __restrict__ wts,
                        int iter, int first) {
    const int j4 = threadIdx.x;
    const int c  = blockIdx.y;
    const int e0 = c * DCdeg;
    const float w = wts[iter];
    const float4* vll = (const float4*)varllr;
    float4*      c2v4 = (float4*)c2v;

    float t[DCdeg][4];
#pragma unroll
    for (int k = 0; k < DCdeg; ++k) {
        int v = Hc[e0 + k];                               // uniform -> s_load
        float4 vl = vll[(size_t)v * RS4 + j4];
        float4 cv;
        if (first) { cv.x = 0.f; cv.y = 0.f; cv.z = 0.f; cv.w = 0.f; }
        else       { cv = c2v4[(size_t)(e0 + k) * RS4 + j4]; }
        t[k][0] = vl.x - cv.x; t[k][1] = vl.y - cv.y;
        t[k][2] = vl.z - cv.z; t[k][3] = vl.w - cv.w;
    }

    float r[DCdeg][4];
#pragma unroll
    for (int comp = 0; comp < 4; ++comp) {                // 4 independent chains
        float s[DCdeg], a[DCdeg];
        float prod = 1.0f;
#pragma unroll
        for (int k = 0; k < DCdeg; ++k) {
            float tk = t[k][comp];
            s[k] = (tk > 0.0f) ? 1.0f : ((tk < 0.0f) ? -1.0f : 0.0f);
            prod *= s[k];
            a[k] = fabsf(tk);
        }
        float amin = a[0]; int imin = 0;
#pragma unroll
        for (int k = 1; k < DCdeg; ++k)
            if (a[k] < amin) { amin = a[k]; imin = k; }   // first argmin (ties)
        float min2 = __builtin_inff();
#pragma unroll
        for (int k = 0; k < DCdeg; ++k)
            if (k != imin && a[k] < min2) min2 = a[k];
#pragma unroll
        for (int k = 0; k < DCdeg; ++k) {
            float os = prod * ((s[k] == 0.0f) ? 1.0f : s[k]); // prod/s == prod*s
            float mo = (k == imin) ? min2 : amin;
            r[k][comp] = w * os * mo;
        }
    }
#pragma unroll
    for (int k = 0; k < DCdeg; ++k) {
        float4 q; q.x = r[k][0]; q.y = r[k][1]; q.z = r[k][2]; q.w = r[k][3];
        c2v4[(size_t)(e0 + k) * RS4 + j4] = q;
    }
}

// -------------------- final aggregation + transpose-out: -> out[B][V] (coalesced)
__launch_bounds__(256)
__global__ void k_final(const float* __restrict__ chT, const float* __restrict__ c2v,
                        const int* __restrict__ v2e, float* __restrict__ out,
                        int b0_global) {
    __shared__ float t2[32 * 33];            // 33-stride: conflict-free both ways
    const int jb0 = blockIdx.x * 32;
    const int v0  = blockIdx.y * 32;
    const int bl  = threadIdx.x & 31;

    for (int vv = threadIdx.x >> 5; vv < 32; vv += 8) {
        int v = v0 + vv, j = jb0 + bl;
        int e0 = v2e[v * 3], e1 = v2e[v * 3 + 1], e2 = v2e[v * 3 + 2];
        float val = chT[(size_t)v * CBATCH + j]
                  + c2v[(size_t)e0 * CBATCH + j]
                  + c2v[(size_t)e1 * CBATCH + j]
                  + c2v[(size_t)e2 * CBATCH + j];
        t2[vv * 33 + bl] = val;
    }
    __syncthreads();
    const int vl = threadIdx.x & 31;
    for (int bb = threadIdx.x >> 5; bb < 32; bb += 8) {
        out[(size_t)(b0_global + jb0 + bb) * NVARS + v0 + vl] = t2[vl * 33 + bb];
    }
}

// ------------------------------------------------------------------ launcher
extern "C" void kernel_launch(void* const* d_in, const int* in_sizes, int n_in,
                              void* d_out, int out_size, void* d_ws, size_t ws_size,
                              hipStream_t stream) {
    (void)in_sizes; (void)n_in; (void)out_size; (void)ws_size;
    const float* ch  = (const float*)d_in[0];
    const float* wts = (const float*)d_in[1];
    const int*   Hc  = (const int*)d_in[3];
    float* out = (float*)d_out;

    char* ws = (char*)d_ws;
    float* chT    = (float*)(ws);                                        // 16 MB
    float* varllr = (float*)(ws + (size_t)NVARS * CBATCH * 4);           // 16 MB
    float* c2v    = (float*)(ws + 2 * (size_t)NVARS * CBATCH * 4);       // 48 MB
    int*   v2e    = (int*)(ws + 2 * (size_t)NVARS * CBATCH * 4
                              + (size_t)NEDGES * CBATCH * 4);
    int*   cnt    = (int*)((char*)v2e + (size_t)NVARS * DVdeg * 4);

    // Build deterministic inverse (var -> 3 edges) map once per launch.
    k_zero_cnt <<<NVARS  / 256, 256, 0, stream>>>(cnt);
    k_build_v2e<<<NEDGES / 256, 256, 0, stream>>>(Hc, cnt, v2e);
    k_sort_v2e <<<NVARS  / 256, 256, 0, stream>>>(v2e);

    for (int chunk = 0; chunk < NBATCH / CBATCH; ++chunk) {
        const int b0 = chunk * CBATCH;
        k_transpose_in<<<dim3(CBATCH / 32, NVARS / 32), 256, 0, stream>>>(ch, chT, b0);
        // Iteration 0: c2v == 0 -> var_llrs == chT; skip the var pass.
        k_check<<<dim3(1, NCHECKS), 128, 0, stream>>>(chT, c2v, Hc, wts, 0, 1);
        for (int it = 1; it < NITER; ++it) {
            k_var  <<<dim3(1, NVARS),   128, 0, stream>>>(chT, c2v, v2e, varllr);
            k_check<<<dim3(1, NCHECKS), 128, 0, stream>>>(varllr, c2v, Hc, wts, it, 0);
        }
        k_final<<<dim3(CBATCH / 32, NVARS / 32), 256, 0, stream>>>(chT, c2v, v2e, out, b0);
    }
}